// SparseLinear_10462540333279
// MI455X (gfx1250) — compile-verified
//
#include <hip/hip_runtime.h>

// Problem constants (match reference).
#define IN_F  4096
#define OUT_F 4096
#define BATCH 128

typedef __attribute__((ext_vector_type(2))) float v2f;
typedef __attribute__((ext_vector_type(4))) float v4f;
typedef __attribute__((ext_vector_type(8))) float v8f;

// ---------------------------------------------------------------------------
// Kernel 1: zero the dense-W scratch (64 MB), vectorized b128 stores.
// ---------------------------------------------------------------------------
__global__ void zero_dense(v4f* __restrict__ W4, int n4) {
    int i = blockIdx.x * blockDim.x + threadIdx.x;
    if (i < n4) {
        W4[i] = (v4f){0.f, 0.f, 0.f, 0.f};
    }
}

// ---------------------------------------------------------------------------
// Kernel 2: COO scatter -> dense W. One fp32 atomic per nonzero (duplicates
// in the random COO indices must sum, matching segment_sum semantics).
// ---------------------------------------------------------------------------
__global__ void coo_scatter(const float* __restrict__ vals,
                            const int*   __restrict__ rows,
                            const int*   __restrict__ cols,
                            float* __restrict__ W, int nnz) {
    int i = blockIdx.x * blockDim.x + threadIdx.x;
    if (i < nnz) {
        size_t idx = (size_t)rows[i] * IN_F + (size_t)cols[i];
        __hip_atomic_fetch_add(&W[idx], vals[i],
                               __ATOMIC_RELAXED, __HIP_MEMORY_SCOPE_AGENT);
    }
}

// ---------------------------------------------------------------------------
// Kernel 3: dense GEMM  out[b][r] = bias[r] + sum_c W[r][c] * inp[b][c]
// via V_WMMA_F32_16X16X4_F32. One wave computes a 16(M) x 64(N) strip.
//
// K is processed 8 at a time ("K8-step"): each lane loads ONE b128 at
// row + 8s + 4*half. Pair .xy feeds a WMMA whose (half,VGPR) slots carry
// K = {8s,8s+1, 8s+4,8s+5}; pair .zw feeds one carrying {8s+2,8s+3, 8s+6,
// 8s+7}. A and B use the identical slot->K permutation and the two WMMAs
// partition K=8s..8s+7, which is all WMMA requires (dot product is
// commutative over K). => 5 b128 loads per 8 WMMAs.
//
// Software pipeline with explicit distance PF=4 K8-steps through rotating
// register buffers: a load sits ~32 WMMAs ahead of its consumer, keeping
// ~20 loads in flight per wave instead of the serialized
// load -> s_wait_loadcnt 0 -> wmma pattern.
// ---------------------------------------------------------------------------
__global__ void __launch_bounds__(32) dense_wmma_gemm(
        const float* __restrict__ W,
        const float* __restrict__ inp,
        const float* __restrict__ bias,
        float* __restrict__ out,
        int kdim) {                     // == IN_F, runtime to bound unrolling
    const int r0   = blockIdx.x * 16;   // 256 M tiles
    const int b0   = blockIdx.y * 64;   // 2 N groups of 64
    const int lane = threadIdx.x & 31;
    const int half = lane >> 4;
    const int l16  = lane & 15;
    const int koff = half * 4;          // disjoint 4-float windows per half

    const float* arow = W   + (size_t)(r0 + l16) * IN_F + koff;
    const float* br0  = inp + (size_t)(b0 +  0 + l16) * IN_F + koff;
    const float* br1  = inp + (size_t)(b0 + 16 + l16) * IN_F + koff;
    const float* br2  = inp + (size_t)(b0 + 32 + l16) * IN_F + koff;
    const float* br3  = inp + (size_t)(b0 + 48 + l16) * IN_F + koff;

    v8f acc0 = {}, acc1 = {}, acc2 = {}, acc3 = {};

    constexpr int PF = 4;               // prefetch distance in K8-steps
    const int steps  = kdim >> 3;       // 512 K8-steps
    const int nmain  = (steps - PF) / PF;   // 127 outer iterations

    v4f A[PF], B0[PF], B1[PF], B2[PF], B3[PF];

    // One K8-step: 8 WMMAs from the staged b128s (pair .xy then .zw).
    auto k8_step = [&](v4f av, v4f b0v, v4f b1v, v4f b2v, v4f b3v) {
        v2f alo = __builtin_shufflevector(av, av, 0, 1);
        v2f ahi = __builtin_shufflevector(av, av, 2, 3);
        v2f l0 = __builtin_shufflevector(b0v, b0v, 0, 1);
        v2f h0 = __builtin_shufflevector(b0v, b0v, 2, 3);
        v2f l1 = __builtin_shufflevector(b1v, b1v, 0, 1);
        v2f h1 = __builtin_shufflevector(b1v, b1v, 2, 3);
        v2f l2 = __builtin_shufflevector(b2v, b2v, 0, 1);
        v2f h2 = __builtin_shufflevector(b2v, b2v, 2, 3);
        v2f l3 = __builtin_shufflevector(b3v, b3v, 0, 1);
        v2f h3 = __builtin_shufflevector(b3v, b3v, 2, 3);
        acc0 = __builtin_amdgcn_wmma_f32_16x16x4_f32(false, alo, false, l0,
                                                     (short)0, acc0, false, false);
        acc1 = __builtin_amdgcn_wmma_f32_16x16x4_f32(false, alo, false, l1,
                                                     (short)0, acc1, false, false);
        acc2 = __builtin_amdgcn_wmma_f32_16x16x4_f32(false, alo, false, l2,
                                                     (short)0, acc2, false, false);
        acc3 = __builtin_amdgcn_wmma_f32_16x16x4_f32(false, alo, false, l3,
                                                     (short)0, acc3, false, false);
        acc0 = __builtin_amdgcn_wmma_f32_16x16x4_f32(false, ahi, false, h0,
                                                     (short)0, acc0, false, false);
        acc1 = __builtin_amdgcn_wmma_f32_16x16x4_f32(false, ahi, false, h1,
                                                     (short)0, acc1, false, false);
        acc2 = __builtin_amdgcn_wmma_f32_16x16x4_f32(false, ahi, false, h2,
                                                     (short)0, acc2, false, false);
        acc3 = __builtin_amdgcn_wmma_f32_16x16x4_f32(false, ahi, false, h3,
                                                     (short)0, acc3, false, false);
    };

    // Prologue: stage K8-steps 0..PF-1.
#pragma unroll
    for (int p = 0; p < PF; ++p) {
        A[p]  = *(const v4f*)(arow + (size_t)p * 8);
        B0[p] = *(const v4f*)(br0  + (size_t)p * 8);
        B1[p] = *(const v4f*)(br1  + (size_t)p * 8);
        B2[p] = *(const v4f*)(br2  + (size_t)p * 8);
        B3[p] = *(const v4f*)(br3  + (size_t)p * 8);
    }

    // Main loop: compute step s = PF*t + i from slot i, refill slot i with
    // step s+PF. Runtime bound prevents complete unroll; inner loop fully
    // unrolls so slot indices are compile-time constants (registers).
    int sbase = 0;
    for (int t = 0; t < nmain; ++t) {
#pragma unroll
        for (int i = 0; i < PF; ++i) {
            v4f av  = A[i];
            v4f b0v = B0[i], b1v = B1[i], b2v = B2[i], b3v = B3[i];

            const size_t off = (size_t)(sbase + i + PF) * 8;
            A[i]  = *(const v4f*)(arow + off);
            B0[i] = *(const v4f*)(br0  + off);
            B1[i] = *(const v4f*)(br1  + off);
            B2[i] = *(const v4f*)(br2  + off);
            B3[i] = *(const v4f*)(br3  + off);

            k8_step(av, b0v, b1v, b2v, b3v);
        }
        sbase += PF;
    }

    // Drain: last PF K8-steps (already staged; nmain*PF ≡ 0 mod PF).
#pragma unroll
    for (int i = 0; i < PF; ++i) {
        k8_step(A[i], B0[i], B1[i], B2[i], B3[i]);
    }

    // Epilogue: bias add + transpose to out[B][OUT_F] via contiguous stores.
    const int mbase = r0 + half * 8;
    v4f bv0 = *(const v4f*)(bias + mbase);
    v4f bv1 = *(const v4f*)(bias + mbase + 4);

    auto store_tile = [&](v8f acc, float* o) {
        v4f lo, hi;
#pragma unroll
        for (int v = 0; v < 4; ++v) {
            lo[v] = acc[v]     + bv0[v];
            hi[v] = acc[v + 4] + bv1[v];
        }
        *(v4f*)o       = lo;
        *(v4f*)(o + 4) = hi;
    };

    store_tile(acc0, out + (size_t)(b0 +  0 + l16) * OUT_F + mbase);
    store_tile(acc1, out + (size_t)(b0 + 16 + l16) * OUT_F + mbase);
    store_tile(acc2, out + (size_t)(b0 + 32 + l16) * OUT_F + mbase);
    store_tile(acc3, out + (size_t)(b0 + 48 + l16) * OUT_F + mbase);
}

// ---------------------------------------------------------------------------
// Launch: zero W -> scatter COO into W -> WMMA GEMM with fused bias+transpose.
// d_ws holds the dense W: OUT_F*IN_F fp32 = 64 MB (L2-resident on MI455X).
// ---------------------------------------------------------------------------
extern "C" void kernel_launch(void* const* d_in, const int* in_sizes, int n_in,
                              void* d_out, int out_size, void* d_ws, size_t ws_size,
                              hipStream_t stream) {
    const float* inp  = (const float*)d_in[0];
    const float* wval = (const float*)d_in[1];
    const int*   wrow = (const int*)d_in[2];
    const int*   wcol = (const int*)d_in[3];
    const float* bias = (const float*)d_in[4];
    float*       out  = (float*)d_out;
    float*       W    = (float*)d_ws;        // OUT_F * IN_F floats
    const int    nnz  = in_sizes[1];

    const int n4 = (OUT_F * IN_F) / 4;
    zero_dense<<<(n4 + 255) / 256, 256, 0, stream>>>((v4f*)W, n4);

    coo_scatter<<<(nnz + 255) / 256, 256, 0, stream>>>(wval, wrow, wcol, W, nnz);

    dim3 grid(OUT_F / 16, BATCH / 64);
    dense_wmma_gemm<<<grid, 32, 0, stream>>>(W, inp, bias, out, IN_F);
}